// BaseSoftMax_79173427134542
// MI455X (gfx1250) — compile-verified
//
#include <hip/hip_runtime.h>

// ---------------- problem constants (from reference) ----------------
#define H 384
#define W 384
#define HWSZ (H * W)          // 147456
#define C_X 64
#define C_PTX 64
#define C_IN 128
#define C_OUT 64
#define N_TGT 120000
#define N_PTS 960000
#define XOUT_ELEMS (C_OUT * HWSZ)   // 9437184 floats (first output)

typedef __attribute__((ext_vector_type(2))) float v2f;
typedef __attribute__((ext_vector_type(8))) float v8f;

// ---------------------------------------------------------------
// 1) gather pixel features: featpx[t, c] = x[c*HW + pixel_tgt_idx[t]]
// ---------------------------------------------------------------
__global__ void k_gather_featpx(const float* __restrict__ x,
                                const int* __restrict__ pix,
                                float* __restrict__ featpx) {
    int idx = blockIdx.x * blockDim.x + threadIdx.x;
    if (idx >= N_TGT * C_X) return;
    int t = idx >> 6;
    int c = idx & 63;
    featpx[idx] = x[(size_t)c * HWSZ + pix[t]];
}

// ---------------------------------------------------------------
// 2) per-point: seg id (binary search on CSR rowptr) + softmax logit
// ---------------------------------------------------------------
__global__ void k_logits_seg(const float* __restrict__ ptx,
                             const float* __restrict__ featpx,
                             const float* __restrict__ sm_w,   // [128]
                             const float* __restrict__ sm_b,   // [1]
                             const int* __restrict__ point_key,
                             int* __restrict__ seg,
                             float* __restrict__ logits) {
    int p = blockIdx.x * blockDim.x + threadIdx.x;
    if (p >= N_PTS) return;
    // binary search: point_key[t] <= p < point_key[t+1]
    int lo = 0, hi = N_TGT;
    while (lo + 1 < hi) {
        int mid = (lo + hi) >> 1;
        if (point_key[mid] <= p) lo = mid; else hi = mid;
    }
    seg[p] = lo;

    float acc = sm_b[0];
    const float4* a = (const float4*)(ptx + (size_t)p * C_PTX);
    const float4* b = (const float4*)(featpx + (size_t)lo * C_X);
    const float4* w0 = (const float4*)(sm_w);
    const float4* w1 = (const float4*)(sm_w + C_PTX);
#pragma unroll
    for (int i = 0; i < C_PTX / 4; ++i) {
        float4 av = a[i], wv = w0[i];
        acc += av.x * wv.x + av.y * wv.y + av.z * wv.z + av.w * wv.w;
    }
#pragma unroll
    for (int i = 0; i < C_X / 4; ++i) {
        float4 bv = b[i], wv = w1[i];
        acc += bv.x * wv.x + bv.y * wv.y + bv.z * wv.z + bv.w * wv.w;
    }
    logits[p] = acc;
}

// ---------------------------------------------------------------
// 3) per-segment softmax weights over its points
// ---------------------------------------------------------------
__global__ void k_weights(const int* __restrict__ point_key,
                          const float* __restrict__ logits,
                          float* __restrict__ weight) {
    int t = blockIdx.x * blockDim.x + threadIdx.x;
    if (t >= N_TGT) return;
    int beg = point_key[t], end = point_key[t + 1];
    float m = -3.4e38f;
    for (int p = beg; p < end; ++p) m = fmaxf(m, logits[p]);
    float s = 0.f;
    for (int p = beg; p < end; ++p) s += __expf(logits[p] - m);
    float inv = 1.f / (s + 1e-16f);
    for (int p = beg; p < end; ++p) weight[p] = __expf(logits[p] - m) * inv;
}

// ---------------------------------------------------------------
// 4) x_out = x (base image; scatter pass overwrites target pixels)
// ---------------------------------------------------------------
__global__ void k_copy_x(const float4* __restrict__ src, float4* __restrict__ dst) {
    int i = blockIdx.x * blockDim.x + threadIdx.x;
    if (i < XOUT_ELEMS / 4) dst[i] = src[i];
}

// ---------------------------------------------------------------
// 5) WMMA GEMM: ptx_out[p,:] = feat[p,:] @ fo_w + fo_b
//    feat[p,:] = [ptx[p, 0:64] | featpx[seg[p], 0:64]]
//    block = 128 threads (4 waves); wave computes a 16x64 tile, K = 128.
// ---------------------------------------------------------------
__global__ void __launch_bounds__(128)
k_gemm_wmma(const float* __restrict__ ptx,
            const float* __restrict__ featpx,
            const int* __restrict__ seg,
            const float* __restrict__ fo_w,   // [128 x 64]
            const float* __restrict__ fo_b,   // [64]
            float* __restrict__ ptx_out) {
    __shared__ float s_w[C_IN * C_OUT];       // 32 KB
    __shared__ float s_feat[4][16][C_IN];     // 32 KB

    const int tid  = threadIdx.x;
    const int lane = tid & 31;
    const int wv   = tid >> 5;

    // stage fo_w in LDS (all waves share)
    for (int i = tid; i < (C_IN * C_OUT) / 4; i += 128)
        ((float4*)s_w)[i] = ((const float4*)fo_w)[i];

    // stage this wave's 16 points x 128 feat in LDS
    const int p0 = blockIdx.x * 64 + wv * 16;
    for (int i = 0; i < 16; ++i) {
        int p = p0 + i;
        if (lane < 16) {
            ((float4*)&s_feat[wv][i][0])[lane] =
                ((const float4*)(ptx + (size_t)p * C_PTX))[lane];
        } else {
            int t = seg[p];
            ((float4*)&s_feat[wv][i][C_PTX])[lane - 16] =
                ((const float4*)(featpx + (size_t)t * C_X))[lane - 16];
        }
    }
    __syncthreads();

    v8f acc[4];
#pragma unroll
    for (int nt = 0; nt < 4; ++nt)
#pragma unroll
        for (int v = 0; v < 8; ++v) acc[nt][v] = 0.f;

    // ISA 16x4 f32 A layout: lanes 0-15 -> K={0,1}; lanes 16-31 -> K={2,3}
    const int mrow  = lane & 15;
    const int khalf = (lane >> 4) * 2;
    const int ncol  = lane & 15;

#pragma unroll 4
    for (int kk = 0; kk < C_IN / 4; ++kk) {
        const int k0 = kk * 4;
        v2f a;
        a.x = s_feat[wv][mrow][k0 + khalf];
        a.y = s_feat[wv][mrow][k0 + khalf + 1];
#pragma unroll
        for (int nt = 0; nt < 4; ++nt) {
            v2f b;
            b.x = s_w[(k0 + khalf) * C_OUT + nt * 16 + ncol];
            b.y = s_w[(k0 + khalf + 1) * C_OUT + nt * 16 + ncol];
            acc[nt] = __builtin_amdgcn_wmma_f32_16x16x4_f32(
                false, a, false, b, (short)0, acc[nt], false, false);
        }
    }

    // C/D layout: VGPR v holds M=v (lanes 0-15) / M=v+8 (lanes 16-31), N=lane%16
    const int mbase = (lane >> 4) * 8;
#pragma unroll
    for (int nt = 0; nt < 4; ++nt) {
        int n = nt * 16 + ncol;
        float bias = fo_b[n];
#pragma unroll
        for (int v = 0; v < 8; ++v) {
            int m = mbase + v;
            ptx_out[(size_t)(p0 + m) * C_OUT + n] = acc[nt][v] + bias;
        }
    }
}

// ---------------------------------------------------------------
// 6) segment-weighted aggregation scattered into x_out (NCHW)
//    one wave per segment; lane covers channels {lane, lane+32}
// ---------------------------------------------------------------
__global__ void k_agg_scatter(const int* __restrict__ point_key,
                              const int* __restrict__ pix,
                              const float* __restrict__ weight,
                              const float* __restrict__ ptx_out,
                              float* __restrict__ xout) {
    int lane = threadIdx.x & 31;
    int wv   = threadIdx.x >> 5;
    int t = blockIdx.x * 8 + wv;
    if (t >= N_TGT) return;
    int beg = point_key[t], end = point_key[t + 1];
    float a0 = 0.f, a1 = 0.f;
    for (int p = beg; p < end; ++p) {
        float wt = weight[p];
        const float* row = ptx_out + (size_t)p * C_OUT;
        a0 += wt * row[lane];
        a1 += wt * row[lane + 32];
    }
    int pixel = pix[t];
    xout[(size_t)lane * HWSZ + pixel] = a0;
    xout[(size_t)(lane + 32) * HWSZ + pixel] = a1;
}

// ---------------------------------------------------------------
extern "C" void kernel_launch(void* const* d_in, const int* in_sizes, int n_in,
                              void* d_out, int out_size, void* d_ws, size_t ws_size,
                              hipStream_t stream) {
    const float* x         = (const float*)d_in[0];
    const float* ptx       = (const float*)d_in[1];
    const float* sm_w      = (const float*)d_in[2];
    const float* sm_b      = (const float*)d_in[3];
    const float* fo_w      = (const float*)d_in[4];
    const float* fo_b      = (const float*)d_in[5];
    const int*   point_key = (const int*)d_in[6];
    const int*   pix       = (const int*)d_in[7];

    float* x_out   = (float*)d_out;
    float* ptx_out = (float*)d_out + XOUT_ELEMS;

    // workspace partition
    char* ws = (char*)d_ws;
    int*   seg    = (int*)(ws);                                   // N_PTS ints
    float* logits = (float*)(ws + (size_t)N_PTS * 4);             // N_PTS f32
    float* weight = (float*)(ws + (size_t)N_PTS * 8);             // N_PTS f32
    float* featpx = (float*)(ws + (size_t)N_PTS * 12);            // N_TGT*64 f32

    k_gather_featpx<<<(N_TGT * C_X + 255) / 256, 256, 0, stream>>>(x, pix, featpx);
    k_logits_seg<<<(N_PTS + 255) / 256, 256, 0, stream>>>(
        ptx, featpx, sm_w, sm_b, point_key, seg, logits);
    k_weights<<<(N_TGT + 255) / 256, 256, 0, stream>>>(point_key, logits, weight);
    k_copy_x<<<(XOUT_ELEMS / 4 + 255) / 256, 256, 0, stream>>>(
        (const float4*)x, (float4*)x_out);
    k_gemm_wmma<<<N_PTS / 64, 128, 0, stream>>>(
        ptx, featpx, seg, fo_w, fo_b, ptx_out);
    k_agg_scatter<<<(N_TGT + 7) / 8, 256, 0, stream>>>(
        point_key, pix, weight, ptx_out, x_out);
}